// TwoStreamReIDNetwork_10728828305723
// MI455X (gfx1250) — compile-verified
//
// MI455X (gfx1250) fused MCB pipeline:
//   count-sketch (LDS ds_add_f32 scatter)  ->  packed complex 8192-pt FFT
//   (radix-16 first step on V_WMMA_F32_16X16X4_F32 + 9 batched Stockham
//    radix-2 stages in LDS)  ->  spectrum unpack & pointwise product  ->
//   inverse hybrid FFT  ->  LayerNorm  ->  L2 normalize  ->  coalesced store.
// One row per 256-thread block; 160 KB LDS => 2 blocks per 320 KB WGP.
#include <hip/hip_runtime.h>
#include <math.h>

typedef __attribute__((ext_vector_type(2))) float v2f;
typedef __attribute__((ext_vector_type(8))) float v8f;

#define BDIM   256
#define DFDIM  8192
#define D1_    2048
#define D2_    1024
#define NHALF  4096
#define LN_EPS 1e-5f
#define L2_EPS 1e-12f
#define PI_F   3.14159265358979323846f

// Hybrid-FFT output layout: natural index k lives at pos(k).
__device__ __forceinline__ int posmap(int k) {
  return ((k & 15) << 9) | (k >> 4);
}

// Radix-16 first step of the 4-step FFT (N = 16*512), done as a complex
// 16x16 x 16x512 matmul on f32 WMMA:
//   U[k1][n2]          = sum_{n1<16} W16^{-csign*n1*k1} * in[n1*512 + n2]
//   out[k1*512 + n2]   = U[k1][n2] * W8192^{-csign*n2*k1}
// csign = +1 forward, -1 inverse. Tables hold forward roots exp(-2pi i t/N).
__device__ __forceinline__ void dft16_wmma(
    const float* __restrict__ sr, const float* __restrict__ si,
    float* __restrict__ dr, float* __restrict__ di,
    const float* __restrict__ twr, const float* __restrict__ twi,
    float csign, int lane, int wave)
{
  const int mrow  = lane & 15;          // A row / B,C column handled by lane
  const int kbase = (lane >> 4) << 1;   // K slots covered by this lane's A regs
  v2f arr[4], aii[4], ain[4];           // Re(W16), Im(W16), -Im(W16)
  #pragma unroll
  for (int kk = 0; kk < 4; ++kk) {
    int K0 = kk * 4 + kbase;
    float s0, c0, s1, c1;
    __sincosf(-csign * (PI_F / 8.0f) * (float)(mrow * K0),       &s0, &c0);
    __sincosf(-csign * (PI_F / 8.0f) * (float)(mrow * (K0 + 1)), &s1, &c1);
    arr[kk].x = c0;  arr[kk].y = c1;
    aii[kk].x = s0;  aii[kk].y = s1;
    ain[kk].x = -s0; ain[kk].y = -s1;   // f32 WMMA has no A-neg modifier
  }
  #pragma unroll
  for (int t = 0; t < 4; ++t) {
    const int tile = wave * 4 + t;      // 8 waves x 4 tiles = 512 columns
    const int n2 = tile * 16 + mrow;
    v8f cr = {0.f, 0.f, 0.f, 0.f, 0.f, 0.f, 0.f, 0.f};
    v8f ci = {0.f, 0.f, 0.f, 0.f, 0.f, 0.f, 0.f, 0.f};
    #pragma unroll
    for (int kk = 0; kk < 4; ++kk) {
      const int K0 = kk * 4 + kbase;
      v2f br, bi;
      br.x = sr[K0 * 512 + n2];       br.y = sr[(K0 + 1) * 512 + n2];
      bi.x = si[K0 * 512 + n2];       bi.y = si[(K0 + 1) * 512 + n2];
      // complex matmul: Cr += Ar*Br - Ai*Bi ; Ci += Ar*Bi + Ai*Br
      cr = __builtin_amdgcn_wmma_f32_16x16x4_f32(false, arr[kk], false, br,
                                                 (short)0, cr, false, false);
      cr = __builtin_amdgcn_wmma_f32_16x16x4_f32(false, ain[kk], false, bi,
                                                 (short)0, cr, false, false);
      ci = __builtin_amdgcn_wmma_f32_16x16x4_f32(false, arr[kk], false, bi,
                                                 (short)0, ci, false, false);
      ci = __builtin_amdgcn_wmma_f32_16x16x4_f32(false, aii[kk], false, br,
                                                 (short)0, ci, false, false);
    }
    // Write back with the inter-stage twiddle W_N^{n2*k1} (n2*k1 <= 7665 < N).
    #pragma unroll
    for (int v = 0; v < 8; ++v) {
      const int k1 = v + ((lane >> 4) << 3);   // C layout: M = vgpr + 8*(lane/16)
      const int tI = n2 * k1;
      float wr, wi;
      if (tI < NHALF) { wr = twr[tI];          wi = twi[tI]; }
      else            { wr = -twr[tI - NHALF]; wi = -twi[tI - NHALF]; }
      wi *= csign;                              // conjugate table for inverse
      const float ur = cr[v], ui = ci[v];
      dr[k1 * 512 + n2] = ur * wr - ui * wi;
      di[k1 * 512 + n2] = ur * wi + ui * wr;
    }
  }
}

// One batched Stockham radix-2 stage: 16 independent 512-point DIF FFTs
// (blocks of 512), 4096 butterflies total, m = 2^mlog, l = 256/m.
__device__ __forceinline__ void stage512(
    const float* __restrict__ sr, const float* __restrict__ si,
    float* __restrict__ dr, float* __restrict__ di,
    const float* __restrict__ twr, const float* __restrict__ twi,
    float csign, int mlog, int tid)
{
  const int m = 1 << mlog;
  for (int t = tid; t < 4096; t += BDIM) {
    const int base = (t >> 8) << 9;     // sub-FFT block * 512
    const int tt = t & 255;
    const int j = tt >> mlog;
    const int k = tt & (m - 1);
    const int i0 = base + k + j * m;
    const int i1 = i0 + 256;
    const int o0 = base + k + 2 * j * m;
    const int o1 = o0 + m;
    const int idx = (j * m) << 4;       // j * (8192/(2*l_block)) = 16*j*m < 4096
    const float wr = twr[idx];
    const float wi = csign * twi[idx];
    const float c0r = sr[i0], c0i = si[i0];
    const float c1r = sr[i1], c1i = si[i1];
    dr[o0] = c0r + c1r;
    di[o0] = c0i + c1i;
    const float ur = c0r - c1r, ui = c0i - c1i;
    dr[o1] = ur * wr - ui * wi;
    di[o1] = ur * wi + ui * wr;
  }
}

extern "C" __global__ void __launch_bounds__(BDIM)
mcb_fused_kernel(const float* __restrict__ x, const float* __restrict__ y,
                 const float* __restrict__ s1, const float* __restrict__ s2,
                 const float* __restrict__ gamma, const float* __restrict__ beta,
                 const int* __restrict__ h1, const int* __restrict__ h2,
                 float* __restrict__ out)
{
  extern __shared__ float smem[];       // 160 KB total -> 2 blocks / WGP
  float* Pr = smem;                     // complex ping-pong pair P
  float* Pi = Pr + DFDIM;
  float* Qr = Pi + DFDIM;               // complex ping-pong pair Q
  float* Qi = Qr + DFDIM;
  float* twr = Qi + DFDIM;              // exp(-2pi i t/8192), t < 4096
  float* twi = twr + NHALF;
  float* red = twr;                     // reduction scratch (aliases dead table)

  const int tid = threadIdx.x;
  const int b = blockIdx.x;
  const int lane = tid & 31;
  const int wave = tid >> 5;

  // --- root-of-unity table + zeroed sketch accumulators ---
  for (int t = tid; t < NHALF; t += BDIM) {
    float s, c;
    __sincosf(-2.0f * PI_F * (float)t * (1.0f / (float)DFDIM), &s, &c);
    twr[t] = c; twi[t] = s;
  }
  for (int t = tid; t < DFDIM; t += BDIM) { Pr[t] = 0.0f; Pi[t] = 0.0f; }
  __syncthreads();

  // --- count sketches: x -> real lane, y -> imag lane (packed complex) ---
  {
    const float* xr = x + (size_t)b * D1_;
    for (int i = tid; i < D1_; i += BDIM)
      atomicAdd(&Pr[h1[i]], xr[i] * s1[i]);     // ds_add_f32
    const float* yr = y + (size_t)b * D2_;
    for (int i = tid; i < D2_; i += BDIM)
      atomicAdd(&Pi[h2[i]], yr[i] * s2[i]);
  }
  __syncthreads();

  // --- forward FFT: WMMA radix-16 step, then 9 Stockham radix-2 stages ---
  dft16_wmma(Pr, Pi, Qr, Qi, twr, twi, +1.0f, lane, wave);
  __syncthreads();
  float *ar = Qr, *ai = Qi, *br = Pr, *bi = Pi;
  for (int s = 0; s < 9; ++s) {
    stage512(ar, ai, br, bi, twr, twi, +1.0f, s, tid);
    __syncthreads();
    float* t0 = ar; ar = br; br = t0;
    float* t1 = ai; ai = bi; bi = t1;
  }
  // packed spectrum Z[k] now in (ar,ai) at posmap(k)

  // --- unpack X,Y spectra and multiply: H[k] into (br,bi), natural order ---
  for (int k = tid; k < DFDIM; k += BDIM) {
    const int km = (DFDIM - k) & (DFDIM - 1);
    const int pk = posmap(k), pm = posmap(km);
    const float a0 = ar[pk], b0 = ai[pk];
    const float c0 = ar[pm], d0 = ai[pm];
    const float Xr = 0.5f * (a0 + c0), Xi = 0.5f * (b0 - d0);
    const float Yr = 0.5f * (b0 + d0), Yi = 0.5f * (c0 - a0);
    br[k] = Xr * Yr - Xi * Yi;
    bi[k] = Xr * Yi + Xi * Yr;
  }
  __syncthreads();

  // --- inverse FFT of H (conjugated twiddles; 1/N folded in later) ---
  dft16_wmma(br, bi, ar, ai, twr, twi, -1.0f, lane, wave);
  __syncthreads();
  for (int s = 0; s < 9; ++s) {
    stage512(ar, ai, br, bi, twr, twi, -1.0f, s, tid);
    __syncthreads();
    float* t0 = ar; ar = br; br = t0;
    float* t1 = ai; ai = bi; bi = t1;
  }
  // fused[n] = ar[posmap(n)] / N  (real part; imag ~ 0 by construction)

  // --- LayerNorm statistics (permutation-invariant over LDS) ---
  const float invN = 1.0f / (float)DFDIM;
  float sum = 0.0f, ssq = 0.0f;
  for (int p = tid; p < DFDIM; p += BDIM) {
    const float v = ar[p] * invN;
    sum += v; ssq += v * v;
  }
  __syncthreads();                 // twiddle table dead from here; reuse as red[]
  red[tid] = sum; red[BDIM + tid] = ssq;
  __syncthreads();
  for (int off = BDIM / 2; off > 0; off >>= 1) {
    if (tid < off) {
      red[tid]        += red[tid + off];
      red[BDIM + tid] += red[BDIM + tid + off];
    }
    __syncthreads();
  }
  const float mu   = red[0] * invN;
  const float var  = red[BDIM] * invN - mu * mu;
  const float rsig = rsqrtf(var + LN_EPS);
  __syncthreads();

  // --- affine normalize, stash normed in br[], accumulate L2 ---
  float acc = 0.0f;
  for (int n = tid; n < DFDIM; n += BDIM) {
    const float f = ar[posmap(n)] * invN;
    const float nv = (f - mu) * rsig * gamma[n] + beta[n];
    br[n] = nv;
    acc += nv * nv;
  }
  __syncthreads();
  red[tid] = acc;
  __syncthreads();
  for (int off = BDIM / 2; off > 0; off >>= 1) {
    if (tid < off) red[tid] += red[tid + off];
    __syncthreads();
  }
  const float rn = 1.0f / fmaxf(sqrtf(red[0]), L2_EPS);

  // --- coalesced store ---
  float* orow = out + (size_t)b * DFDIM;
  for (int n = tid; n < DFDIM; n += BDIM)
    orow[n] = br[n] * rn;
}

extern "C" void kernel_launch(void* const* d_in, const int* in_sizes, int n_in,
                              void* d_out, int out_size, void* d_ws, size_t ws_size,
                              hipStream_t stream) {
  const float* x     = (const float*)d_in[0];
  const float* y     = (const float*)d_in[1];
  const float* s1    = (const float*)d_in[2];
  const float* s2    = (const float*)d_in[3];
  const float* gamma = (const float*)d_in[4];
  const float* beta  = (const float*)d_in[5];
  const int*   h1    = (const int*)d_in[6];
  const int*   h2    = (const int*)d_in[7];
  float* out = (float*)d_out;

  const int B = in_sizes[0] / D1_;                    // 4096 rows
  const size_t smem = (size_t)(4 * DFDIM + 2 * NHALF) * sizeof(float);  // 160 KB
  mcb_fused_kernel<<<B, BDIM, smem, stream>>>(x, y, s1, s2, gamma, beta,
                                              h1, h2, out);
}